// LorentzNotGroupNorm_21199958573144
// MI455X (gfx1250) — compile-verified
//
#include <hip/hip_runtime.h>
#include <hip/hip_bf16.h>
#include <cstdint>

// Lorentz group-norm on MI455X (gfx1250).
// Bandwidth-bound: >=206MB streamed per launch -> ~9us roofline at 23.3TB/s.
// Strategy: per 16-pixel tile (2 groups of 8), compute the 16x16 Euclidean
// Gram matrix with chained V_WMMA_F32_16X16X4_F32 (A==B, K=128). All Lorentz
// inner products derive from the Gram (+ channel-0 corrections), so the whole
// log/transport math reduces to per-row scalars; only the final per-channel
// apply touches the full vectors again. Tile staging uses the CDNA5 async
// global->LDS pipe (ASYNCcnt) when available.

typedef __attribute__((ext_vector_type(2))) float v2f;
typedef __attribute__((ext_vector_type(8))) float v8f;
typedef __attribute__((ext_vector_type(4))) int   v4i;

#define AS1 __attribute__((address_space(1)))
#define AS3 __attribute__((address_space(3)))

constexpr int BS  = 64;
constexpr int HW  = 3136;              // 56*56
constexpr int C   = 128;
constexpr int GRP = 8;                 // pixels per group (groups=8 input)
constexpr int TILE_PIX = 16;           // 2 groups per WMMA tile
constexpr int TILES_PER_BATCH  = HW / TILE_PIX;          // 196
constexpr int WAVES_PER_BLOCK  = 4;                      // 128 threads
constexpr int BLOCKS_PER_BATCH = TILES_PER_BATCH / WAVES_PER_BLOCK; // 49
constexpr int NBLOCKS = BS * BLOCKS_PER_BATCH;           // 3136
constexpr int PITCH = 132;             // LDS row pitch (floats): bank-conflict-free WMMA feed
constexpr float KEPS      = 1e-5f;
constexpr float MINN      = 1e-8f;
constexpr float MAX_EUC   = 10.0f;
constexpr float INV_SPG   = 1.0f / 392.0f;   // 1 / (HW/GRP)

#if __has_builtin(__builtin_amdgcn_global_load_async_to_lds_b128)
#define USE_ASYNC_LDS 1
#else
#define USE_ASYNC_LDS 0
#endif

__device__ __forceinline__ void async_wait0() {
#if __has_builtin(__builtin_amdgcn_s_wait_asynccnt)
  __builtin_amdgcn_s_wait_asynccnt(0);
#else
  asm volatile("s_wait_asynccnt 0" ::: "memory");
#endif
}

__device__ __forceinline__ float seg_red8(float v) {
  v += __shfl_xor(v, 1, 32);
  v += __shfl_xor(v, 2, 32);
  v += __shfl_xor(v, 4, 32);
  return v;
}

struct TileScalars { float f[8], gp[8], cf[8], nrm[8]; };

// Per-wave tile math. Valid results: lanes 0-7 hold rows 0-7 (group A),
// lanes 24-31 hold rows 8-15 (group B). Other lanes compute harmless garbage.
__device__ __forceinline__ void tile_scalars(const float* __restrict__ tl,
                                             int lane, TileScalars& ts) {
  const int rowA = lane & 15;
  const int off  = (lane < 16) ? 0 : 2;
  const float* bp = tl + rowA * PITCH + off;

  v8f acc = {0.f, 0.f, 0.f, 0.f, 0.f, 0.f, 0.f, 0.f};
#if __has_builtin(__builtin_amdgcn_wmma_f32_16x16x4_f32)
  #pragma unroll
  for (int kc = 0; kc < C / 4; ++kc) {
    v2f a = *(const v2f*)(bp + kc * 4);
    // G = X * X^T : A and B layouts coincide for this operand pair.
    acc = __builtin_amdgcn_wmma_f32_16x16x4_f32(false, a, false, a,
                                                (short)0, acc, false, false);
  }
#else
  // VALU fallback (keeps the round compiling if the builtin signature differs)
  #pragma unroll
  for (int k = 0; k < 8; ++k) {
    const int Mr = k + ((lane < 16) ? 0 : 8);
    const int Nc = lane & 15;
    float s = 0.f;
    for (int c = 0; c < C; ++c) s += tl[Mr * PITCH + c] * tl[Nc * PITCH + c];
    acc[k] = s;
  }
#endif

  const float x0col = tl[rowA * PITCH];     // x0 of row (lane&15)
  const float x0s   = seg_red8(x0col);      // sum of x0 over this segment's group
  const float avg0  = x0s * 0.125f;
  const int   rsel  = (lane < 16) ? 0 : 8;

  float rs[8], sL[8], Gd[8], x0r[8];
  float tE = 0.f, tL = 0.f;
  #pragma unroll
  for (int k = 0; k < 8; ++k) {
    float r  = seg_red8(acc[k]);                       // E row-sum within group
    float xr = __shfl(x0col, k + rsel, 32);            // x0 of this row
    float l  = r - 2.0f * xr * x0s;                    // Lorentz row-sum
    Gd[k]  = __shfl(acc[k], (lane < 16) ? k : (24 + k), 32);  // E[p][p]
    rs[k] = r; sL[k] = l; x0r[k] = xr;
    tE += r; tL += l;
  }
  const float negaa  = -tL * (1.0f / 64.0f);           // -<avg,avg>_L
  const float denomc = sqrtf(fmaxf(negaa, MINN));
  const float idc    = 1.0f / denomc;
  const float mm     = (tL * (1.0f / 64.0f)) * idc * idc;   // <mean,mean>_L
  const float m0     = avg0 * idc;
  const float Ea     = tE * (1.0f / 64.0f);            // <avg,avg>_E

  #pragma unroll
  for (int k = 0; k < 8; ++k) {
    float xmL   = (sL[k] * 0.125f) * idc;              // <x_p,mean>_L
    float alpha = fmaxf(-xmL, 1.0f + 1e-7f);
    float dist  = acoshf(alpha);
    float Lpp   = Gd[k] - 2.0f * x0r[k] * x0r[k];
    float nn    = Lpp - 2.0f * alpha * xmL + alpha * alpha * mm;
    float dn2   = sqrtf(fmaxf(nn, MINN));
    float f     = dist / dn2;                          // x_T = f*x_p + g*avg
    float g     = -(f * alpha) * idc;
    float v0    = f * x0r[k] + g * avg0;               // tangent time component
    float cf    = -v0 / (1.0f + m0);                   // transp0back coefficient
    float gp    = g + cf * idc;                        // x_T' = f*x + gp*avg + cf*o
    ts.f[k] = f; ts.gp[k] = gp; ts.cf[k] = cf;
    float Epa = rs[k] * 0.125f;
    float n2  = f * f * Gd[k] + gp * gp * Ea + cf * cf
              + 2.0f * f * gp * Epa + 2.0f * f * cf * x0r[k]
              + 2.0f * gp * cf * avg0;                 // ||x_T'||_E^2
    ts.nrm[k] = sqrtf(fmaxf(n2, 0.0f));
  }
}

__device__ __forceinline__ void load_tile(const float* __restrict__ x,
                                          size_t gbase, int lane,
                                          float* __restrict__ tl) {
#if USE_ASYNC_LDS
  // CDNA5 async global->LDS DMA: 32 lanes x 16B per op = one 512B row,
  // no VGPR round trip; tracked with ASYNCcnt.
  #pragma unroll
  for (int r = 0; r < TILE_PIX; ++r) {
    const float* g = x + gbase + (size_t)r * C + lane * 4;
    float*       l = tl + r * PITCH + lane * 4;
    __builtin_amdgcn_global_load_async_to_lds_b128(
        (AS1 v4i*)(uintptr_t)g,
        (AS3 v4i*)(uintptr_t)l,
        0 /*offset*/, 0 /*cpol*/);
  }
  async_wait0();
#else
  #pragma unroll
  for (int r = 0; r < TILE_PIX; ++r) {
    float4 v = *(const float4*)(x + gbase + (size_t)r * C + lane * 4);
    *(float4*)(tl + r * PITCH + lane * 4) = v;
  }
#endif
}

// ---------------- Pass 1: per-pixel tangent norms -> per-block partials ----
__global__ __launch_bounds__(128) void lgn_pass1(const float* __restrict__ x,
                                                 float* __restrict__ partial) {
  __shared__ float tileS[WAVES_PER_BLOCK][TILE_PIX * PITCH];
  __shared__ float wn[WAVES_PER_BLOCK][TILE_PIX];
  const int tid = threadIdx.x, w = tid >> 5, lane = tid & 31;
  const int b = blockIdx.x / BLOCKS_PER_BATCH;
  const int t = (blockIdx.x % BLOCKS_PER_BATCH) * WAVES_PER_BLOCK + w;
  const size_t gbase = ((size_t)b * HW + (size_t)t * TILE_PIX) * C;

  float* tl = tileS[w];
  load_tile(x, gbase, lane, tl);
  __syncthreads();

  TileScalars ts;
  tile_scalars(tl, lane, ts);

  if (lane == 0) {
    #pragma unroll
    for (int k = 0; k < 8; ++k) wn[w][k] = ts.nrm[k];
  }
  if (lane == 24) {
    #pragma unroll
    for (int k = 0; k < 8; ++k) wn[w][k + 8] = ts.nrm[k];
  }
  __syncthreads();
  if (tid < GRP) {   // deterministic fixed-order block reduction
    float s = 0.f;
    #pragma unroll
    for (int ww = 0; ww < WAVES_PER_BLOCK; ++ww)
      s += wn[ww][tid] + wn[ww][tid + 8];
    partial[(size_t)blockIdx.x * GRP + tid] = s;
  }
}

// ---------------- Pass 1b: fold 49 block partials per (batch, j) -----------
__global__ __launch_bounds__(512) void lgn_pass1b(const float* __restrict__ partial,
                                                  float* __restrict__ varsum) {
  const int i = threadIdx.x;          // 512 = 64 batches * 8 positions
  const int b = i >> 3, j = i & 7;
  float s = 0.f;
  for (int blk = 0; blk < BLOCKS_PER_BATCH; ++blk)
    s += partial[((size_t)(b * BLOCKS_PER_BATCH + blk)) * GRP + j];
  varsum[i] = s;
}

// ---------------- Pass 2: recompute coefficients, apply, expmap ------------
__global__ __launch_bounds__(128) void lgn_pass2(const float* __restrict__ x,
                                                 const float* __restrict__ gamma,
                                                 const float* __restrict__ beta,
                                                 const float* __restrict__ varsum,
                                                 float* __restrict__ out) {
  __shared__ float tileS[WAVES_PER_BLOCK][TILE_PIX * PITCH];
  __shared__ float scalS[WAVES_PER_BLOCK][TILE_PIX][4];
  const int tid = threadIdx.x, w = tid >> 5, lane = tid & 31;
  const int b = blockIdx.x / BLOCKS_PER_BATCH;
  const int t = (blockIdx.x % BLOCKS_PER_BATCH) * WAVES_PER_BLOCK + w;
  const size_t gbase = ((size_t)b * HW + (size_t)t * TILE_PIX) * C;

  float* tl = tileS[w];
  load_tile(x, gbase, lane, tl);
  __syncthreads();

  TileScalars ts;
  tile_scalars(tl, lane, ts);
  if (lane == 0) {
    #pragma unroll
    for (int k = 0; k < 8; ++k) {
      scalS[w][k][0] = ts.f[k]; scalS[w][k][1] = ts.gp[k]; scalS[w][k][2] = ts.cf[k];
    }
  }
  if (lane == 24) {
    #pragma unroll
    for (int k = 0; k < 8; ++k) {
      scalS[w][k + 8][0] = ts.f[k]; scalS[w][k + 8][1] = ts.gp[k]; scalS[w][k + 8][2] = ts.cf[k];
    }
  }
  __syncthreads();

  // channel layout: lane owns channels [4*lane, 4*lane+3]
  const float4 g4 = *(const float4*)(gamma + lane * 4);
  const float4 b4 = *(const float4*)(beta  + lane * 4);
  float bpart = b4.x * b4.x + b4.y * b4.y + b4.z * b4.z + b4.w * b4.w;
  if (lane == 0) bpart -= 2.0f * b4.x * b4.x;        // Lorentz sign on ch 0
  #pragma unroll
  for (int m = 16; m >= 1; m >>= 1) bpart += __shfl_xor(bpart, m, 32);
  const float bb    = bpart;                          // <beta,beta>_L
  const float beta0 = __shfl(b4.x, 0, 32);
  const float boo   = bb - 2.0f * beta0 - 1.0f;       // <beta+o,beta+o>_L

  float4 avA, avB;
  {
    float ax = 0, ay = 0, az = 0, aw = 0;
    #pragma unroll
    for (int r = 0; r < 8; ++r) {
      float4 v = *(const float4*)(tl + r * PITCH + lane * 4);
      ax += v.x; ay += v.y; az += v.z; aw += v.w;
    }
    avA = make_float4(ax * 0.125f, ay * 0.125f, az * 0.125f, aw * 0.125f);
    ax = ay = az = aw = 0;
    #pragma unroll
    for (int r = 8; r < 16; ++r) {
      float4 v = *(const float4*)(tl + r * PITCH + lane * 4);
      ax += v.x; ay += v.y; az += v.z; aw += v.w;
    }
    avB = make_float4(ax * 0.125f, ay * 0.125f, az * 0.125f, aw * 0.125f);
  }

  #pragma unroll 4
  for (int p = 0; p < TILE_PIX; ++p) {
    const float f  = scalS[w][p][0];
    const float gp = scalS[w][p][1];
    const float cf = scalS[w][p][2];
    const float vj = varsum[b * GRP + (p & 7)] * INV_SPG;
    const float iv = 1.0f / (vj + KEPS);
    float4 xv = *(const float4*)(tl + p * PITCH + lane * 4);
    float4 av = (p < 8) ? avA : avB;
    float ux = f * xv.x + gp * av.x + ((lane == 0) ? cf : 0.0f);
    float uy = f * xv.y + gp * av.y;
    float uz = f * xv.z + gp * av.z;
    float uw = f * xv.w + gp * av.w;
    ux *= g4.x * iv; uy *= g4.y * iv; uz *= g4.z * iv; uw *= g4.w * iv;

    float pa = ux * ux + uy * uy + uz * uz + uw * uw;      // ||u||^2 partial
    float pl = b4.x * ux + b4.y * uy + b4.z * uz + b4.w * uw;
    if (lane == 0) pl -= 2.0f * b4.x * ux;                 // <beta,u>_L partial
    #pragma unroll
    for (int m = 16; m >= 1; m >>= 1) {
      pa += __shfl_xor(pa, m, 32);
      pl += __shfl_xor(pl, m, 32);
    }
    const float u0 = __shfl(ux, 0, 32);
    const float nr = sqrtf(pa);
    const float sc = fminf(1.0f, MAX_EUC / fmaxf(nr, MINN));   // rescale cap
    const float tt = (pl * sc) / (1.0f + beta0);               // transp0 coeff
    // v' = sc*u + tt*(beta + o);  ||v'||_L^2 in closed form:
    float n2 = sc * sc * (pa - 2.0f * u0 * u0)
             + 2.0f * tt * sc * (pl - u0) + tt * tt * boo;
    float nrm = sqrtf(fmaxf(n2, MINN));
    float sh = sinhf(nrm) / nrm, chh = coshf(nrm);
    float ox = chh * b4.x + sh * (sc * ux + tt * b4.x + ((lane == 0) ? tt : 0.0f));
    float oy = chh * b4.y + sh * (sc * uy + tt * b4.y);
    float oz = chh * b4.z + sh * (sc * uz + tt * b4.z);
    float ow = chh * b4.w + sh * (sc * uw + tt * b4.w);
    *(float4*)(out + gbase + (size_t)p * C + lane * 4) = make_float4(ox, oy, oz, ow);
  }
}

extern "C" void kernel_launch(void* const* d_in, const int* in_sizes, int n_in,
                              void* d_out, int out_size, void* d_ws, size_t ws_size,
                              hipStream_t stream) {
  const float* x     = (const float*)d_in[0];
  const float* gamma = (const float*)d_in[1];
  const float* beta  = (const float*)d_in[2];
  // d_in[3] = groups (==8), baked into the tiling at compile time.
  float* out     = (float*)d_out;
  float* partial = (float*)d_ws;                       // NBLOCKS*8 floats
  float* varsum  = partial + (size_t)NBLOCKS * GRP;    // 512 floats

  lgn_pass1 <<<NBLOCKS, 128, 0, stream>>>(x, partial);
  lgn_pass1b<<<1, 512, 0, stream>>>(partial, varsum);
  lgn_pass2 <<<NBLOCKS, 128, 0, stream>>>(x, gamma, beta, varsum, out);
}